// SimpleSTLClassifier_9079560864408
// MI455X (gfx1250) — compile-verified
//
#include <hip/hip_runtime.h>
#include <hip/hip_bf16.h>

typedef __attribute__((ext_vector_type(2))) float v2f;
typedef __attribute__((ext_vector_type(8))) float v8f;

#define DIN 64
#define HDIM 64
#define LPAD 65   // LDS row stride (65 mod 64 == 1 -> conflict-free column reads)

// Guaranteed-native fp32 global atomic add (non-returning, device scope).
// Avoids any chance of clang expanding atomicrmw fadd into a CAS loop.
__device__ __forceinline__ void atom_add_f32(float* p, float v) {
  asm volatile("global_atomic_add_f32 %0, %1, off scope:SCOPE_DEV"
               :: "v"(p), "v"(v)
               : "memory");
}

// ---------------------------------------------------------------------------
// h_lin = X @ W   (rows x 64) @ (64 x 64), fp32 WMMA 16x16x4, K-loop of 16.
// Block = 128 thr = 4 waves. Block owns 16 output rows; wave w owns cols 16w..16w+15.
// ---------------------------------------------------------------------------
__global__ void __launch_bounds__(128)
gcn_gemm_wmma(const float* __restrict__ X, const float* __restrict__ W,
              float* __restrict__ T, int rows) {
  __shared__ float xs[16 * LPAD];   // X tile, 16 x 64
  __shared__ float wsh[64 * LPAD];  // full W,  64 x 64

  const int tid = threadIdx.x;
  const int m0  = blockIdx.x * 16;

  // stage X tile (zero-fill OOB rows so WMMA math is safe)
  for (int i = tid; i < 16 * 64; i += 128) {
    const int r = i >> 6, c = i & 63;
    const int gr = m0 + r;
    xs[r * LPAD + c] = (gr < rows) ? X[(size_t)gr * 64 + c] : 0.0f;
  }
  // stage W (64x64 = 16KB)
  for (int i = tid; i < 64 * 64; i += 128) {
    const int r = i >> 6, c = i & 63;
    wsh[r * LPAD + c] = W[i];
  }
  __syncthreads();

  const int lane = tid & 31;
  const int wv   = tid >> 5;
  const int n0   = wv * 16;
  const int m    = lane & 15;   // row (A) / col (B,C) within tile
  const int half = lane >> 4;   // lane half selects K pairs

  v8f c = {};
  for (int kk = 0; kk < 16; ++kk) {
    const int k0 = kk * 4;
    v2f a, b;
    // A 16x4 f32: VGPR j, lanes[half]: K = k0 + 2*half + j, M = m
    a[0] = xs[m * LPAD + k0 + 2 * half + 0];
    a[1] = xs[m * LPAD + k0 + 2 * half + 1];
    // B 4x16 f32: VGPR j, lanes[half]: K = k0 + 2*half + j, N = n0 + m
    b[0] = wsh[(k0 + 2 * half + 0) * LPAD + n0 + m];
    b[1] = wsh[(k0 + 2 * half + 1) * LPAD + n0 + m];
    c = __builtin_amdgcn_wmma_f32_16x16x4_f32(false, a, false, b,
                                              (short)0, c, false, false);
  }

  // C/D 16x16 f32: VGPR j, lanes[half]: M = j + 8*half, N = n0 + m
  for (int j = 0; j < 8; ++j) {
    const int gr = m0 + j + 8 * half;
    if (gr < rows) T[(size_t)gr * 64 + n0 + m] = c[j];
  }
}

// ---------------------------------------------------------------------------
// helpers
// ---------------------------------------------------------------------------
__global__ void zero_f32_v4(float4* __restrict__ p, int n4) {
  int i = blockIdx.x * blockDim.x + threadIdx.x;
  if (i < n4) p[i] = make_float4(0.f, 0.f, 0.f, 0.f);
}

__global__ void init_deg(float* __restrict__ deg, int n) {
  int i = blockIdx.x * blockDim.x + threadIdx.x;
  if (i < n) deg[i] = 1.0f;   // self-loop contribution
}

__global__ void accum_deg(const int* __restrict__ dst, float* __restrict__ deg,
                          int e) {
  int i = blockIdx.x * blockDim.x + threadIdx.x;
  if (i < e) atom_add_f32(&deg[dst[i]], 1.0f);
}

__global__ void rsqrt_inplace(float* __restrict__ d, int n) {
  int i = blockIdx.x * blockDim.x + threadIdx.x;
  if (i < n) d[i] = rsqrtf(d[i]);   // deg >= 1 always (self loops)
}

// one wave32 per edge: float2 gather from t[src], 2 fp32 atomics into agg[dst]
__global__ void __launch_bounds__(256)
scatter_edges(const float* __restrict__ t, const int* __restrict__ src,
              const int* __restrict__ dst, const float* __restrict__ dinv,
              float* __restrict__ agg, int e) {
  const int wid  = (blockIdx.x * 256 + threadIdx.x) >> 5;
  const int lane = threadIdx.x & 31;
  if (wid >= e) return;
  const int s = src[wid];
  const int d = dst[wid];
  const float nrm = dinv[s] * dinv[d];
  const float2 v = ((const float2*)(t + (size_t)s * 64))[lane];
  float* ap = agg + (size_t)d * 64 + lane * 2;
  atom_add_f32(ap + 0, v.x * nrm);
  atom_add_f32(ap + 1, v.y * nrm);
}

// h = relu(agg + t*dinv^2 (self loop) + bias), in place on agg; float4 lanes
__global__ void finalize_layer(float* __restrict__ agg, const float* __restrict__ t,
                               const float* __restrict__ dinv,
                               const float* __restrict__ bias, int n) {
  int idx = blockIdx.x * blockDim.x + threadIdx.x;   // over n*16 float4 groups
  if (idx >= n * 16) return;
  const int i = idx >> 4;
  const float di = dinv[i];
  const float sc = di * di;
  float4 a        = ((float4*)agg)[idx];
  const float4 tv = ((const float4*)t)[idx];
  const float4 bv = ((const float4*)bias)[idx & 15];
  a.x = fmaxf(a.x + tv.x * sc + bv.x, 0.0f);
  a.y = fmaxf(a.y + tv.y * sc + bv.y, 0.0f);
  a.z = fmaxf(a.z + tv.z * sc + bv.z, 0.0f);
  a.w = fmaxf(a.w + tv.w * sc + bv.w, 0.0f);
  ((float4*)agg)[idx] = a;
}

// one wave32 per node: atomic accumulate into per-graph sums + counts
__global__ void __launch_bounds__(256)
pool_nodes(const float* __restrict__ h, const int* __restrict__ batch,
           float* __restrict__ sums, float* __restrict__ cnt, int n) {
  const int wid  = (blockIdx.x * 256 + threadIdx.x) >> 5;
  const int lane = threadIdx.x & 31;
  if (wid >= n) return;
  const int g = batch[wid];
  const float2 v = ((const float2*)(h + (size_t)wid * 64))[lane];
  float* sp = sums + (size_t)g * 64 + lane * 2;
  atom_add_f32(sp + 0, v.x);
  atom_add_f32(sp + 1, v.y);
  if (lane == 0) atom_add_f32(cnt + g, 1.0f);
}

// tiny classifier: pooled = sums/cnt; z = relu(pooled@Wc1+bc1); out = z@Wc2+bc2
__global__ void __launch_bounds__(256)
classifier(const float* __restrict__ sums, const float* __restrict__ cnt,
           const float* __restrict__ Wc1, const float* __restrict__ bc1,
           const float* __restrict__ Wc2, const float* __restrict__ bc2,
           float* __restrict__ out, int G) {
  __shared__ float pooled[64 * 64];
  __shared__ float z[64 * 32];
  const int tid = threadIdx.x;
  for (int i = tid; i < G * 64; i += 256) {
    const int g = i >> 6;
    pooled[i] = sums[i] / fmaxf(cnt[g], 1.0f);
  }
  __syncthreads();
  for (int i = tid; i < G * 32; i += 256) {
    const int g = i >> 5, j = i & 31;
    float acc = bc1[j];
    for (int k = 0; k < 64; ++k) acc += pooled[g * 64 + k] * Wc1[k * 32 + j];
    z[i] = fmaxf(acc, 0.0f);
  }
  __syncthreads();
  for (int i = tid; i < G * 2; i += 256) {
    const int g = i >> 1, c = i & 1;
    float acc = bc2[c];
    for (int k = 0; k < 32; ++k) acc += z[g * 32 + k] * Wc2[k * 2 + c];
    out[i] = acc;
  }
}

// ---------------------------------------------------------------------------
extern "C" void kernel_launch(void* const* d_in, const int* in_sizes, int n_in,
                              void* d_out, int out_size, void* d_ws, size_t ws_size,
                              hipStream_t stream) {
  const float* x   = (const float*)d_in[0];
  const float* W1  = (const float*)d_in[1];
  const float* b1  = (const float*)d_in[2];
  const float* W2  = (const float*)d_in[3];
  const float* b2  = (const float*)d_in[4];
  const float* W3  = (const float*)d_in[5];
  const float* b3  = (const float*)d_in[6];
  const float* Wc1 = (const float*)d_in[7];
  const float* bc1 = (const float*)d_in[8];
  const float* Wc2 = (const float*)d_in[9];
  const float* bc2 = (const float*)d_in[10];
  const int*   ei  = (const int*)d_in[11];
  const int*   bat = (const int*)d_in[12];
  float* out = (float*)d_out;

  const int N = in_sizes[0] / DIN;
  const int E = in_sizes[11] / 2;
  const int G = out_size / 2;
  const int* esrc = ei;
  const int* edst = ei + E;

  // workspace layout (floats)
  float* ws   = (float*)d_ws;
  float* dinv = ws;                         // N
  float* t    = dinv + N;                   // N*64
  float* hA   = t + (size_t)N * 64;         // N*64
  float* hB   = hA + (size_t)N * 64;        // N*64
  float* sums = hB + (size_t)N * 64;        // G*64
  float* cnt  = sums + (size_t)G * 64;      // G
  (void)ws_size; (void)n_in;

  const int B = 256;
  const int nNodeBlk = (N + B - 1) / B;
  const int nEdgeBlk = (E + B - 1) / B;
  const int nVec4Blk = (N * 16 + B - 1) / B;   // N*64 floats as float4
  const int nEdgeWaveBlk = (E + 7) / 8;        // 8 waves (edges) per 256-thr block
  const int nNodeWaveBlk = (N + 7) / 8;
  const int nGemmBlk = (N + 15) / 16;

  // degree -> dinv
  init_deg<<<nNodeBlk, B, 0, stream>>>(dinv, N);
  accum_deg<<<nEdgeBlk, B, 0, stream>>>(edst, dinv, E);
  rsqrt_inplace<<<nNodeBlk, B, 0, stream>>>(dinv, N);

  // layer 1: x -> hA
  gcn_gemm_wmma<<<nGemmBlk, 128, 0, stream>>>(x, W1, t, N);
  zero_f32_v4<<<nVec4Blk, B, 0, stream>>>((float4*)hA, N * 16);
  scatter_edges<<<nEdgeWaveBlk, B, 0, stream>>>(t, esrc, edst, dinv, hA, E);
  finalize_layer<<<nVec4Blk, B, 0, stream>>>(hA, t, dinv, b1, N);

  // layer 2: hA -> hB
  gcn_gemm_wmma<<<nGemmBlk, 128, 0, stream>>>(hA, W2, t, N);
  zero_f32_v4<<<nVec4Blk, B, 0, stream>>>((float4*)hB, N * 16);
  scatter_edges<<<nEdgeWaveBlk, B, 0, stream>>>(t, esrc, edst, dinv, hB, E);
  finalize_layer<<<nVec4Blk, B, 0, stream>>>(hB, t, dinv, b2, N);

  // layer 3: hB -> hA
  gcn_gemm_wmma<<<nGemmBlk, 128, 0, stream>>>(hB, W3, t, N);
  zero_f32_v4<<<nVec4Blk, B, 0, stream>>>((float4*)hA, N * 16);
  scatter_edges<<<nEdgeWaveBlk, B, 0, stream>>>(t, esrc, edst, dinv, hA, E);
  finalize_layer<<<nVec4Blk, B, 0, stream>>>(hA, t, dinv, b3, N);

  // pool + classify
  zero_f32_v4<<<((G * 64 + G) / 4 + B - 1) / B, B, 0, stream>>>((float4*)sums,
                                                                (G * 64 + G) / 4);
  pool_nodes<<<nNodeWaveBlk, B, 0, stream>>>(hA, bat, sums, cnt, N);
  classifier<<<1, B, 0, stream>>>(sums, cnt, Wc1, bc1, Wc2, bc2, out, G);
}